// CrossEntropy_25134148616683
// MI455X (gfx1250) — compile-verified
//
#include <hip/hip_runtime.h>
#include <stdint.h>

// ---------------------------------------------------------------------------
// BCE cost + top-k membership for [T=64, B=128, V=4880].
// Memory-bound: 320 MB of f32 reads -> ~14us floor @ 23.3 TB/s on MI455X.
// No GEMM structure -> WMMA inapplicable; CDNA5-specific path used instead:
// global_load_async_to_lds_b128 (ASYNCcnt) to stage each y_hat row in LDS,
// since the row is re-read for rank computation (candidate filtering).
// ---------------------------------------------------------------------------

#define T_DIM 64
#define B_DIM 128
#define V_DIM 4880
#define ROWS (T_DIM * B_DIM)      // 8192
#define NF4 (V_DIM / 4)           // 1220 float4 per row
#define CAND_CAP 160
#define POS_CAP 256
#define CUTOFF 0.986885f          // 1 - 64/4880: E[#candidates]=64 >> 20
#define LOG_EPS 1e-8f

typedef float v4f __attribute__((ext_vector_type(4)));

__global__ __launch_bounds__(256) void bce_topk_rows(
    const float* __restrict__ yhat, const float* __restrict__ y,
    const float* __restrict__ len, float* __restrict__ ws_ce,
    unsigned int* __restrict__ ws_cnt)
{
  __shared__ __align__(16) float sh[V_DIM];        // y_hat row (19520 B)
  __shared__ float    cand_val[CAND_CAP];
  __shared__ unsigned cand_tag[CAND_CAP];          // idx | (ispos << 31)
  __shared__ unsigned pos_idx[POS_CAP];            // fallback path only
  __shared__ unsigned cnt_cand, cnt_pos, tp[4];
  __shared__ float    redbuf[8];

  const int tid = threadIdx.x;
  const int row = blockIdx.x;                      // row = t*B + b
  const size_t base = (size_t)row * V_DIM;

  if (tid == 0) { cnt_cand = 0; cnt_pos = 0; tp[0] = tp[1] = tp[2] = tp[3] = 0; }

  // ---- async stage y_hat row into LDS (CDNA5 ASYNCcnt path) ----
  for (int i = tid; i < NF4; i += 256) {
    unsigned lds = (unsigned)(uintptr_t)(&sh[i * 4]);
    const float* g = yhat + base + (size_t)i * 4;
    asm volatile("global_load_async_to_lds_b128 %0, %1, off"
                 :: "v"(lds), "v"(g) : "memory");
  }
  asm volatile("s_wait_asynccnt 0x0" ::: "memory");
  __syncthreads();

  // ---- single streaming pass: BCE + candidate/positive collection ----
  const v4f* y4 = (const v4f*)(y + base);
  float ce = 0.0f;
  for (int i = tid; i < NF4; i += 256) {
    v4f yv = __builtin_nontemporal_load(&y4[i]);
    v4f xv = *((const v4f*)sh + i);
    #pragma unroll
    for (int j = 0; j < 4; ++j) {
      float x  = xv[j];
      float yy = yv[j];
      bool pos = (yy != 0.0f);
      ce -= pos ? __logf(x + LOG_EPS) : __logf(1.0f - x + LOG_EPS);
      unsigned idx = (unsigned)(i * 4 + j);
      if (x > CUTOFF) {
        unsigned s = atomicAdd(&cnt_cand, 1u);
        if (s < CAND_CAP) { cand_val[s] = x; cand_tag[s] = idx | (pos ? 0x80000000u : 0u); }
      }
      if (pos) {
        unsigned s = atomicAdd(&cnt_pos, 1u);
        if (s < POS_CAP) pos_idx[s] = idx;
      }
    }
  }

  // ---- block-reduce CE (wave32 shuffles, fixed order -> deterministic) ----
  for (int m = 16; m; m >>= 1) ce += __shfl_xor(ce, m, 32);
  if ((tid & 31) == 0) redbuf[tid >> 5] = ce;
  __syncthreads();
  if (tid < 32) {
    float v = (tid < 8) ? redbuf[tid] : 0.0f;
    for (int m = 4; m; m >>= 1) v += __shfl_xor(v, m, 32);
    if (tid == 0) ws_ce[row] = v / len[row % B_DIM];
  }
  __syncthreads();

  const unsigned ncand = cnt_cand;
  const unsigned npos  = cnt_pos;

  if (ncand >= 20 && ncand <= CAND_CAP) {
    // Fast path: any element with rank < 20 is above CUTOFF (>=20 candidates
    // outrank everything below it). Rank of a positive candidate is computed
    // against the candidate list only (all greater/equal values are in it).
    for (unsigned j = tid; j < ncand; j += 256) {
      unsigned tag = cand_tag[j];
      if (tag & 0x80000000u) {
        float v = cand_val[j];
        unsigned idx = tag & 0x7FFFFFFFu;
        unsigned gt = 0, eqb = 0;
        for (unsigned m = 0; m < ncand; ++m) {
          float vm = cand_val[m];
          unsigned im = cand_tag[m] & 0x7FFFFFFFu;
          gt  += (vm > v);
          eqb += (vm == v) && (im < idx);   // stable-argsort tie-break
        }
        unsigned rank = gt + eqb;
        if (rank < 5)  atomicAdd(&tp[0], 1u);
        if (rank < 10) atomicAdd(&tp[1], 1u);
        if (rank < 15) atomicAdd(&tp[2], 1u);
        if (rank < 20) atomicAdd(&tp[3], 1u);
      }
    }
  } else {
    // Rare deterministic fallback: exact full-row scan per positive.
    unsigned np = npos < POS_CAP ? npos : POS_CAP;
    int wave = tid >> 5, lane = tid & 31;
    for (unsigned p = (unsigned)wave; p < np; p += 8) {
      unsigned pi = pos_idx[p];
      float pv = sh[pi];
      unsigned gt = 0, eqb = 0;
      for (int i = lane; i < V_DIM; i += 32) {
        float x = sh[i];
        gt  += (x > pv);
        eqb += (x == pv) && ((unsigned)i < pi);
      }
      unsigned packed = (gt << 13) | eqb;          // eqb < 8192, no carry
      for (int m = 16; m; m >>= 1) packed += __shfl_xor(packed, m, 32);
      if (lane == 0) {
        unsigned rank = (packed >> 13) + (packed & 8191u);
        if (rank < 5)  atomicAdd(&tp[0], 1u);
        if (rank < 10) atomicAdd(&tp[1], 1u);
        if (rank < 15) atomicAdd(&tp[2], 1u);
        if (rank < 20) atomicAdd(&tp[3], 1u);
      }
    }
  }
  __syncthreads();

  if (tid == 0) {
    if (tp[0]) atomicAdd(&ws_cnt[0], tp[0]);
    if (tp[1]) atomicAdd(&ws_cnt[1], tp[1]);
    if (tp[2]) atomicAdd(&ws_cnt[2], tp[2]);
    if (tp[3]) atomicAdd(&ws_cnt[3], tp[3]);
    atomicAdd(&ws_cnt[4], npos);                   // n_pos total
  }
}

__global__ __launch_bounds__(256) void bce_topk_final(
    const float* __restrict__ ws_ce, const unsigned int* __restrict__ ws_cnt,
    float* __restrict__ out)
{
  __shared__ float redbuf[8];
  const int tid = threadIdx.x;
  float s = 0.0f;
  for (int i = tid; i < ROWS; i += 256) s += ws_ce[i];   // fixed order
  for (int m = 16; m; m >>= 1) s += __shfl_xor(s, m, 32);
  if ((tid & 31) == 0) redbuf[tid >> 5] = s;
  __syncthreads();
  if (tid < 32) {
    float v = (tid < 8) ? redbuf[tid] : 0.0f;
    for (int m = 4; m; m >>= 1) v += __shfl_xor(v, m, 32);
    if (tid == 0) out[0] = v / (float)B_DIM;             // cost = mean over B
  }
  if (tid < 4) {
    out[1 + tid] = (float)ws_cnt[tid];                   // tps
    out[5 + tid] = (float)ws_cnt[4];                     // totals = n_pos
  }
}

extern "C" void kernel_launch(void* const* d_in, const int* in_sizes, int n_in,
                              void* d_out, int out_size, void* d_ws, size_t ws_size,
                              hipStream_t stream) {
  (void)in_sizes; (void)n_in; (void)out_size; (void)ws_size;
  const float* yhat = (const float*)d_in[0];
  const float* y    = (const float*)d_in[1];
  const float* len  = (const float*)d_in[2];
  float* out = (float*)d_out;
  float* ws_ce = (float*)d_ws;                                      // 8192 f32
  unsigned int* ws_cnt = (unsigned int*)((char*)d_ws + ROWS * sizeof(float)); // 5 u32

  hipMemsetAsync(ws_cnt, 0, 5 * sizeof(unsigned int), stream);
  bce_topk_rows<<<ROWS, 256, 0, stream>>>(yhat, y, len, ws_ce, ws_cnt);
  bce_topk_final<<<1, 256, 0, stream>>>(ws_ce, ws_cnt, out);
}